// LocallyConnectedNN_77137612636309
// MI455X (gfx1250) — compile-verified
//
#include <hip/hip_runtime.h>
#include <hip/hip_bf16.h>

typedef float v2f __attribute__((ext_vector_type(2)));
typedef float v4f __attribute__((ext_vector_type(4)));
typedef float v8f __attribute__((ext_vector_type(8)));

#define BN_EPS 1e-5f

// D = A(16x4 f32) * B(4x16 f32) + C(16x16 f32), wave32 CDNA5 WMMA.
__device__ __forceinline__ v8f wmma_f32x4(v2f a, v2f b, v8f c) {
  return __builtin_amdgcn_wmma_f32_16x16x4_f32(false, a, false, b, (short)0, c,
                                               false, false);
}

// ---------------------------------------------------------------------------
// conv1: [B,1,16,16] -> z1 [B,16,14,14] (raw, pre-BN) + per-channel sum/sumsq
// K=9 is too small for WMMA; VALU FMA, memory-bound layer.
// ---------------------------------------------------------------------------
__global__ void __launch_bounds__(256) conv1_kernel(
    const float* __restrict__ x, const float* __restrict__ w1,
    float* __restrict__ z1, double* __restrict__ stats1) {
  __shared__ float wsh[144];
  __shared__ float lsum[16], lsq[16];
  int tid = threadIdx.x;
  if (tid < 144) wsh[tid] = w1[tid];
  if (tid < 16) lsum[tid] = 0.f;
  else if (tid < 32) lsq[tid - 16] = 0.f;
  __syncthreads();
  float ps[16], pq[16];
#pragma unroll
  for (int c = 0; c < 16; ++c) { ps[c] = 0.f; pq[c] = 0.f; }
  int base = blockIdx.x * 256 + tid;
  for (int g = 0; g < 4; ++g) {
    int gp = base + g * 802816;              // pixel id, 16384*196 total
    int b = gp / 196;
    int r = gp % 196;                        // r = yy*14+xx
    int yy = r / 14, xx = r % 14;
    const float* xi = x + b * 256 + yy * 16 + xx;
    float i0 = xi[0],  i1 = xi[1],  i2 = xi[2];
    float i3 = xi[16], i4 = xi[17], i5 = xi[18];
    float i6 = xi[32], i7 = xi[33], i8 = xi[34];
    float* zo = z1 + b * 3136 + r;           // + co*196
#pragma unroll
    for (int co = 0; co < 16; ++co) {
      const float* w = &wsh[co * 9];
      float a = i0 * w[0] + i1 * w[1] + i2 * w[2] + i3 * w[3] + i4 * w[4] +
                i5 * w[5] + i6 * w[6] + i7 * w[7] + i8 * w[8];
      zo[co * 196] = a;
      ps[co] += a;
      pq[co] += a * a;
    }
  }
#pragma unroll
  for (int c = 0; c < 16; ++c) {
    atomicAdd(&lsum[c], ps[c]);
    atomicAdd(&lsq[c], pq[c]);
  }
  __syncthreads();
  if (tid < 16) unsafeAtomicAdd(&stats1[tid], (double)lsum[tid]);
  else if (tid < 32) unsafeAtomicAdd(&stats1[tid], (double)lsq[tid - 16]);
}

// ---------------------------------------------------------------------------
// Fold BN params: st[2c]=gamma*rsqrt(var+eps), st[2c+1]=beta-mean*st[2c]
// ---------------------------------------------------------------------------
__global__ void bn_finalize(const double* __restrict__ stats,
                            const float* __restrict__ g,
                            const float* __restrict__ b, double invN, int C,
                            float* __restrict__ st) {
  int c = threadIdx.x;
  if (c < C) {
    double mean = stats[c] * invN;
    double var = stats[C + c] * invN - mean * mean;
    float s = g[c] * rsqrtf((float)var + BN_EPS);
    float t = b[c] - (float)mean * s;
    st[2 * c] = s;
    st[2 * c + 1] = t;
  }
}

// ---------------------------------------------------------------------------
// conv2: implicit GEMM, M=pixels(16/wave-tile), N=32 co, K=64 (ci,dy,dx).
// k = ci*4 + dy*2 + dx  =>  per K-step t: ci==t (uniform), A pairs contiguous.
// Reads z1 with BN1+ReLU folded; writes z2 [p][32]; accumulates stats2.
// ---------------------------------------------------------------------------
__global__ void __launch_bounds__(128) conv2_kernel(
    const float* __restrict__ z1, const float* __restrict__ w2,
    const float* __restrict__ st1, float* __restrict__ z2,
    double* __restrict__ stats2) {
  __shared__ float Bl[2048];   // Bl[pi*64 + co*2 + lo] = w2[co*64 + pi*2 + lo]
  __shared__ float sst[64];
  int tid = threadIdx.x;
  for (int i = tid; i < 2048; i += 128) {
    int pi = i >> 6, rr = i & 63, co = rr >> 1, lo = rr & 1;
    Bl[i] = w2[co * 64 + pi * 2 + lo];
  }
  if (tid < 64) sst[tid] = 0.f;
  float ss[16], tt[16];
#pragma unroll
  for (int t = 0; t < 16; ++t) { ss[t] = st1[2 * t]; tt[t] = st1[2 * t + 1]; }
  __syncthreads();
  int lane = tid & 31, wv = tid >> 5;
  int ln = lane & 15, lh = lane >> 4;
  int gwave = blockIdx.x * 4 + wv;
  float s0a = 0.f, q0a = 0.f, s1a = 0.f, q1a = 0.f;
  for (int ti = 0; ti < 8; ++ti) {
    int tile = gwave * 8 + ti;
    int p = tile * 16 + ln;
    int b = p / 169;
    int r = p % 169;
    int yy = r / 13, xx = r % 13;
    const float* ap = z1 + b * 3136 + yy * 14 + xx + lh * 14;
    v8f c0 = {}, c1 = {};
#pragma unroll
    for (int t = 0; t < 16; ++t) {
      float a0 = ap[0], a1 = ap[1];
      ap += 196;
      float s = ss[t], tb = tt[t];
      a0 = fmaxf(fmaf(a0, s, tb), 0.f);
      a1 = fmaxf(fmaf(a1, s, tb), 0.f);
      v2f A = {a0, a1};
      const float* bp = &Bl[(2 * t + lh) * 64 + ln * 2];
      v2f B0 = *(const v2f*)bp;
      v2f B1 = *(const v2f*)(bp + 32);
      c0 = wmma_f32x4(A, B0, c0);
      c1 = wmma_f32x4(A, B1, c1);
    }
    int pbase = tile * 16 + lh * 8;
#pragma unroll
    for (int rr2 = 0; rr2 < 8; ++rr2) {
      float v0 = c0[rr2], v1 = c1[rr2];
      z2[(pbase + rr2) * 32 + ln] = v0;
      z2[(pbase + rr2) * 32 + 16 + ln] = v1;
      s0a += v0; q0a += v0 * v0;
      s1a += v1; q1a += v1 * v1;
    }
  }
  atomicAdd(&sst[ln], s0a);
  atomicAdd(&sst[16 + ln], s1a);
  atomicAdd(&sst[32 + ln], q0a);
  atomicAdd(&sst[48 + ln], q1a);
  __syncthreads();
  if (tid < 64) unsafeAtomicAdd(&stats2[tid], (double)sst[tid]);
}

// ---------------------------------------------------------------------------
// conv3 (1x1): GEMM M=pixels, N=64, K=32. Reads z2 with BN2+ReLU folded,
// writes z3 [p][64]; accumulates stats3.
// ---------------------------------------------------------------------------
__global__ void __launch_bounds__(128) conv3_kernel(
    const float* __restrict__ z2, const float* __restrict__ w3,
    const float* __restrict__ st2, float* __restrict__ z3,
    double* __restrict__ stats3) {
  __shared__ float Bl[2048];  // Bl[pi*128 + co*2 + lo] = w3[co*32 + pi*2 + lo]
  __shared__ float stl[64];
  __shared__ float sst[128];
  int tid = threadIdx.x;
  for (int i = tid; i < 2048; i += 128) {
    int pi = i >> 7, rr = i & 127, co = rr >> 1, lo = rr & 1;
    Bl[i] = w3[co * 32 + pi * 2 + lo];
  }
  if (tid < 64) stl[tid] = st2[tid];
  sst[tid] = 0.f;
  __syncthreads();
  int lane = tid & 31, wv = tid >> 5;
  int ln = lane & 15, lh = lane >> 4;
  int gwave = blockIdx.x * 4 + wv;
  float sa[4] = {0.f, 0.f, 0.f, 0.f}, qa[4] = {0.f, 0.f, 0.f, 0.f};
  for (int ti = 0; ti < 8; ++ti) {
    int tile = gwave * 8 + ti;
    int p = tile * 16 + ln;
    const float* ap = z2 + p * 32 + lh * 2;
    v8f c[4] = {};
#pragma unroll
    for (int t = 0; t < 8; ++t) {
      float a0 = ap[4 * t], a1 = ap[4 * t + 1];
      v4f stq = *(const v4f*)&stl[(4 * t + 2 * lh) * 2];
      a0 = fmaxf(fmaf(a0, stq[0], stq[1]), 0.f);
      a1 = fmaxf(fmaf(a1, stq[2], stq[3]), 0.f);
      v2f A = {a0, a1};
      const float* bp = &Bl[(2 * t + lh) * 128 + ln * 2];
#pragma unroll
      for (int n = 0; n < 4; ++n) {
        v2f Bv = *(const v2f*)(bp + n * 32);
        c[n] = wmma_f32x4(A, Bv, c[n]);
      }
    }
    int pbase = tile * 16 + lh * 8;
#pragma unroll
    for (int rr2 = 0; rr2 < 8; ++rr2) {
#pragma unroll
      for (int n = 0; n < 4; ++n) {
        float v = c[n][rr2];
        z3[(pbase + rr2) * 64 + n * 16 + ln] = v;
        sa[n] += v;
        qa[n] += v * v;
      }
    }
  }
#pragma unroll
  for (int n = 0; n < 4; ++n) {
    atomicAdd(&sst[n * 16 + ln], sa[n]);
    atomicAdd(&sst[64 + n * 16 + ln], qa[n]);
  }
  __syncthreads();
  unsafeAtomicAdd(&stats3[tid], (double)sst[tid]);
}

// ---------------------------------------------------------------------------
// Reorder fc_w [10][10816] (flatten index c*169+p) into WMMA-friendly pair
// layout: fr[pi*32 + j*2 + lo] = fc_w[j][c*169+p], k = p*64+c = pi*2+lo,
// zero-padded to N=16. 692 KB, L2-resident.
// ---------------------------------------------------------------------------
__global__ void fcw_reorder_kernel(const float* __restrict__ fc_w,
                                   float* __restrict__ fr) {
  int idx = blockIdx.x * 256 + threadIdx.x;  // exactly 173056 threads
  int pi = idx >> 5, rr = idx & 31, j = rr >> 1, lo = rr & 1;
  int k = pi * 2 + lo;
  int p = k >> 6, c = k & 63;
  fr[idx] = (j < 10) ? fc_w[j * 10816 + c * 169 + p] : 0.f;
}

// ---------------------------------------------------------------------------
// FC: M=16 batch rows per block, N=16 (10 used), K=10816 split across 4 waves.
// Reads z3 with BN3+ReLU folded; LDS combine of partial accumulators.
// ---------------------------------------------------------------------------
__global__ void __launch_bounds__(128) fc_kernel(
    const float* __restrict__ z3, const float* __restrict__ fr,
    const float* __restrict__ st3, const float* __restrict__ fc_b,
    float* __restrict__ out) {
  __shared__ float stl[128];
  __shared__ float cred[128 * 8];
  int tid = threadIdx.x;
  stl[tid] = st3[tid];
  __syncthreads();
  int lane = tid & 31, wv = tid >> 5;
  int ln = lane & 15, lh = lane >> 4;
  int b0 = blockIdx.x * 16;
  const float* ap = z3 + (b0 + ln) * 10816 + lh * 2;
  v8f c = {};
  int t0 = wv * 676, t1 = t0 + 676;
  for (int t = t0; t < t1; ++t) {
    __builtin_prefetch(ap + 4 * t + 256, 0, 0);
    float a0 = ap[4 * t], a1 = ap[4 * t + 1];
    v4f stq = *(const v4f*)&stl[((4 * t + 2 * lh) & 63) * 2];
    a0 = fmaxf(fmaf(a0, stq[0], stq[1]), 0.f);
    a1 = fmaxf(fmaf(a1, stq[2], stq[3]), 0.f);
    v2f A = {a0, a1};
    v2f Bv = *(const v2f*)&fr[(2 * t + lh) * 32 + ln * 2];
    c = wmma_f32x4(A, Bv, c);
  }
#pragma unroll
  for (int r = 0; r < 8; ++r) cred[tid * 8 + r] = c[r];
  __syncthreads();
  if (wv == 0) {
#pragma unroll
    for (int r = 0; r < 8; ++r) {
      float s = cred[lane * 8 + r] + cred[(lane + 32) * 8 + r] +
                cred[(lane + 64) * 8 + r] + cred[(lane + 96) * 8 + r];
      if (ln < 10) out[(b0 + lh * 8 + r) * 10 + ln] = s + fc_b[ln];
    }
  }
}

extern "C" void kernel_launch(void* const* d_in, const int* in_sizes, int n_in,
                              void* d_out, int out_size, void* d_ws,
                              size_t ws_size, hipStream_t stream) {
  (void)in_sizes; (void)n_in; (void)out_size; (void)ws_size;
  const float* x   = (const float*)d_in[0];
  const float* w1  = (const float*)d_in[1];
  const float* w2  = (const float*)d_in[2];
  const float* w3  = (const float*)d_in[3];
  const float* g1  = (const float*)d_in[4];
  const float* b1  = (const float*)d_in[5];
  const float* g2  = (const float*)d_in[6];
  const float* b2  = (const float*)d_in[7];
  const float* g3  = (const float*)d_in[8];
  const float* b3  = (const float*)d_in[9];
  const float* fcw = (const float*)d_in[10];
  const float* fcb = (const float*)d_in[11];
  float* out = (float*)d_out;

  char* ws = (char*)d_ws;
  size_t o = 0;
  float* z1 = (float*)(ws + o); o += (size_t)16384 * 16 * 196 * 4;  // 205.5 MB
  float* z2 = (float*)(ws + o); o += (size_t)16384 * 169 * 32 * 4;  // 354.4 MB
  float* z3 = (float*)(ws + o); o += (size_t)16384 * 169 * 64 * 4;  // 708.8 MB
  float* fr = (float*)(ws + o); o += (size_t)173056 * 4;            // 692 KB
  double* stats1 = (double*)(ws + o); o += 32 * 8;
  double* stats2 = (double*)(ws + o); o += 64 * 8;
  double* stats3 = (double*)(ws + o); o += 128 * 8;
  float* st1 = (float*)(ws + o); o += 32 * 4;
  float* st2 = (float*)(ws + o); o += 64 * 4;
  float* st3 = (float*)(ws + o); o += 128 * 4;

  hipMemsetAsync(stats1, 0, (32 + 64 + 128) * 8, stream);
  fcw_reorder_kernel<<<676, 256, 0, stream>>>(fcw, fr);
  conv1_kernel<<<3136, 256, 0, stream>>>(x, w1, z1, stats1);
  bn_finalize<<<1, 16, 0, stream>>>(stats1, g1, b1, 1.0 / 3211264.0, 16, st1);
  conv2_kernel<<<5408, 128, 0, stream>>>(z1, w2, st1, z2, stats2);
  bn_finalize<<<1, 32, 0, stream>>>(stats2, g2, b2, 1.0 / 2768896.0, 32, st2);
  conv3_kernel<<<5408, 128, 0, stream>>>(z2, w3, st2, z3, stats3);
  bn_finalize<<<1, 64, 0, stream>>>(stats3, g3, b3, 1.0 / 2768896.0, 64, st3);
  fc_kernel<<<1024, 128, 0, stream>>>(z3, fr, st3, fcb, out);
}